// LPF_86835648790832
// MI455X (gfx1250) — compile-verified
//
#include <hip/hip_runtime.h>
#include <math.h>

// ---------------------------------------------------------------------------
// EMA recurrence  y_t = b*x_t + (1-b)*y_{t-1}  over (B=32, T=4096, U=512) f32.
// Bandwidth-bound (0.5 flop/byte, ~768 MB total traffic -> ~33 us at 23.3TB/s).
// 3-phase chunked scan: tails -> carries -> final. Both streaming passes use a
// double-buffered GLOBAL_LOAD_ASYNC_TO_LDS_B128 pipeline (ASYNCcnt-tracked,
// per-wave waits, no block barrier needed since each wave consumes only the
// LDS cells it loaded itself).
// ---------------------------------------------------------------------------

#define EB   32            // batch
#define ET   4096          // time
#define EU   512           // channels
#define ENCH 32            // chunks along T
#define ELL  (ET / ENCH)   // 128 timesteps per chunk
#define ES   8             // timesteps per async stage
#define EK   (ELL / ES)    // 16 stages per chunk

typedef float v4f __attribute__((ext_vector_type(4)));
typedef int   v4i __attribute__((ext_vector_type(4)));

#if __has_builtin(__builtin_amdgcn_global_load_async_to_lds_b128) && \
    __has_builtin(__builtin_amdgcn_s_wait_asynccnt)
#define EMA_HAVE_ASYNC 1
#else
#define EMA_HAVE_ASYNC 0
#endif

// Builtin signature (verified via compiler diagnostics):
//   void __builtin_amdgcn_global_load_async_to_lds_b128(
//       int4 addrspace(1)* src, int4 addrspace(3)* dst, imm int off, imm cpol)
__device__ __forceinline__ __attribute__((address_space(1))) v4i*
as_global_v4(const void* p) {
  return (__attribute__((address_space(1))) v4i*)(p);
}
__device__ __forceinline__ __attribute__((address_space(3))) v4i*
as_lds_v4(void* p) {
  return (__attribute__((address_space(3))) v4i*)(p);
}

// ---------------------------------------------------------------------------
// Pass 1: local chunk-end values (state entering chunk assumed 0).
// Block = 128 threads = one (b, chunk); thread i owns channels [4i, 4i+4).
// Async-staged through LDS, compute-only (single tail store at the end).
// ---------------------------------------------------------------------------
__global__ __launch_bounds__(128) void ema_tails_kernel(
    const float* __restrict__ x, const float* __restrict__ bc,
    float* __restrict__ tails) {
  const int u4    = threadIdx.x;            // 0..127
  const int u     = u4 * 4;
  const int chunk = blockIdx.x & (ENCH - 1);
  const int b     = blockIdx.x >> 5;

  const v4f bv = *(const v4f*)(bc + u);
  const v4f av = 1.0f - bv;

  const float* xp = x + ((size_t)b * ET + (size_t)chunk * ELL) * EU + u;
  v4f y = {0.f, 0.f, 0.f, 0.f};

#if EMA_HAVE_ASYNC
  __shared__ v4f buf[2][ES][EU / 4];        // 32 KB

#pragma unroll
  for (int t = 0; t < ES; ++t) {
    __builtin_amdgcn_global_load_async_to_lds_b128(
        as_global_v4(xp + (size_t)t * EU), as_lds_v4(&buf[0][t][u4]), 0, 0);
  }
  for (int k = 0; k < EK - 1; ++k) {
    const int cur = k & 1;
    const int nxt = cur ^ 1;
#pragma unroll
    for (int t = 0; t < ES; ++t) {
      __builtin_amdgcn_global_load_async_to_lds_b128(
          as_global_v4(xp + (size_t)(k + 1) * ES * EU + (size_t)t * EU),
          as_lds_v4(&buf[nxt][t][u4]), 0, 0);
    }
    __builtin_amdgcn_s_wait_asynccnt(ES);   // stage k complete (in-order)
    asm volatile("" ::: "memory");
#pragma unroll
    for (int t = 0; t < ES; ++t) {
      y = __builtin_elementwise_fma(av, y, bv * buf[cur][t][u4]);
    }
  }
  __builtin_amdgcn_s_wait_asynccnt(0);
  asm volatile("" ::: "memory");
#pragma unroll
  for (int t = 0; t < ES; ++t) {
    y = __builtin_elementwise_fma(av, y, bv * buf[(EK - 1) & 1][t][u4]);
  }
#else
#pragma unroll 8
  for (int t = 0; t < ELL; ++t) {
    const v4f xv = __builtin_nontemporal_load((const v4f*)xp);
    y = __builtin_elementwise_fma(av, y, bv * xv);
    xp += EU;
  }
#endif
  *(v4f*)(tails + (size_t)(b * ENCH + chunk) * EU + u) = y;
}

// ---------------------------------------------------------------------------
// Pass 2: per-channel prefix over chunk tails using a^ELL (7 squarings).
// carries[b][j][u] = state ENTERING chunk j.
// ---------------------------------------------------------------------------
__global__ __launch_bounds__(256) void ema_carries_kernel(
    const float* __restrict__ bc, const float* __restrict__ tails,
    float* __restrict__ carries) {
  const int tid = blockIdx.x * blockDim.x + threadIdx.x;  // B * U/4 = 4096
  const int u   = (tid & (EU / 4 - 1)) * 4;
  const int b   = tid >> 7;

  const v4f bv = *(const v4f*)(bc + u);
  v4f aL = 1.0f - bv;
#pragma unroll
  for (int i = 0; i < 7; ++i) aL *= aL;  // a^(2^7) = a^128 = a^ELL

  v4f carry = {0.f, 0.f, 0.f, 0.f};
#pragma unroll
  for (int j = 0; j < ENCH; ++j) {
    const size_t off = (size_t)(b * ENCH + j) * EU + u;
    *(v4f*)(carries + off) = carry;
    const v4f e = *(const v4f*)(tails + off);
    carry = __builtin_elementwise_fma(aL, carry, e);
  }
}

// ---------------------------------------------------------------------------
// Pass 3: rescan each chunk from its carry, writing y. Same async pipeline,
// plus nontemporal b128 output stores (output never re-read; keep L2 clean).
// ---------------------------------------------------------------------------
__global__ __launch_bounds__(128) void ema_final_kernel(
    const float* __restrict__ x, const float* __restrict__ bc,
    const float* __restrict__ carries, float* __restrict__ out) {
  const int u4    = threadIdx.x;            // 0..127
  const int u     = u4 * 4;
  const int chunk = blockIdx.x & (ENCH - 1);
  const int b     = blockIdx.x >> 5;

  const v4f bv = *(const v4f*)(bc + u);
  const v4f av = 1.0f - bv;

  const size_t base = ((size_t)b * ET + (size_t)chunk * ELL) * EU + u;
  const float* xp = x + base;
  float*       yp = out + base;

  v4f y = *(const v4f*)(carries + (size_t)(b * ENCH + chunk) * EU + u);

#if EMA_HAVE_ASYNC
  __shared__ v4f buf[2][ES][EU / 4];        // 32 KB

#pragma unroll
  for (int t = 0; t < ES; ++t) {
    __builtin_amdgcn_global_load_async_to_lds_b128(
        as_global_v4(xp + (size_t)t * EU), as_lds_v4(&buf[0][t][u4]), 0, 0);
  }
  for (int k = 0; k < EK - 1; ++k) {
    const int cur = k & 1;
    const int nxt = cur ^ 1;
#pragma unroll
    for (int t = 0; t < ES; ++t) {
      __builtin_amdgcn_global_load_async_to_lds_b128(
          as_global_v4(xp + (size_t)(k + 1) * ES * EU + (size_t)t * EU),
          as_lds_v4(&buf[nxt][t][u4]), 0, 0);
    }
    __builtin_amdgcn_s_wait_asynccnt(ES);   // stage k complete (in-order)
    asm volatile("" ::: "memory");
#pragma unroll
    for (int t = 0; t < ES; ++t) {
      y = __builtin_elementwise_fma(av, y, bv * buf[cur][t][u4]);
      __builtin_nontemporal_store(y, (v4f*)(yp + (size_t)(k * ES + t) * EU));
    }
  }
  __builtin_amdgcn_s_wait_asynccnt(0);
  asm volatile("" ::: "memory");
#pragma unroll
  for (int t = 0; t < ES; ++t) {
    y = __builtin_elementwise_fma(av, y, bv * buf[(EK - 1) & 1][t][u4]);
    __builtin_nontemporal_store(
        y, (v4f*)(yp + (size_t)((EK - 1) * ES + t) * EU));
  }
#else
#pragma unroll 8
  for (int t = 0; t < ELL; ++t) {
    const v4f xv = __builtin_nontemporal_load((const v4f*)xp);
    y = __builtin_elementwise_fma(av, y, bv * xv);
    __builtin_nontemporal_store(y, (v4f*)yp);
    xp += EU;
    yp += EU;
  }
#endif
}

extern "C" void kernel_launch(void* const* d_in, const int* in_sizes, int n_in,
                              void* d_out, int out_size, void* d_ws, size_t ws_size,
                              hipStream_t stream) {
  const float* x  = (const float*)d_in[0];   // (B, T, U) f32
  const float* bc = (const float*)d_in[1];   // (U,) f32
  float* y        = (float*)d_out;           // (B, T, U) f32

  float* tails   = (float*)d_ws;                        // 2 MiB
  float* carries = tails + (size_t)EB * ENCH * EU;      // 2 MiB

  const int nCarry = EB * (EU / 4);          // 4096 threads

  ema_tails_kernel<<<EB * ENCH, 128, 0, stream>>>(x, bc, tails);
  ema_carries_kernel<<<nCarry / 256, 256, 0, stream>>>(bc, tails, carries);
  ema_final_kernel<<<EB * ENCH, 128, 0, stream>>>(x, bc, carries, y);
}